// TopKPool_24824910970968
// MI455X (gfx1250) — compile-verified
//
#include <hip/hip_runtime.h>
#include <hip/hip_bf16.h>

// ---------------------------------------------------------------------------
// TopKPool on MI455X (gfx1250, wave32)
//   A_pooled = (A@A)[idx][:,idx]  ==  A[idx,:] @ A[:,idx]   (34.4 GFLOP f32)
//   computed with V_WMMA_F32_16X16X4_F32, 64x64 tile per wave (4x4 frags).
//   B side is pre-gathered+transposed (BT[c,k] = A[k,idx[c]]) when workspace
//   allows, so every fragment load is a row-contiguous global_load_b64.
// ---------------------------------------------------------------------------

#define N_NODES 4096
#define F_DIM   256
#define K_KEEP  2048

typedef __attribute__((ext_vector_type(2))) float v2f;
typedef __attribute__((ext_vector_type(8))) float v8f;

// ---- kernel 0: inverse L2 norm of the (256,1) projection kernel ------------
__global__ void tkp_norm_kernel(const float* __restrict__ kern,
                                float* __restrict__ invnorm) {
    __shared__ float red[256];
    float v = kern[threadIdx.x];
    red[threadIdx.x] = v * v;
    __syncthreads();
    for (int s = 128; s > 0; s >>= 1) {
        if (threadIdx.x < s) red[threadIdx.x] += red[threadIdx.x + s];
        __syncthreads();
    }
    if (threadIdx.x == 0) invnorm[0] = rsqrtf(red[0]);
}

// ---- kernel 1: y = X @ kn, th = tanh(y); one wave per row ------------------
__global__ void tkp_score_kernel(const float* __restrict__ X,
                                 const float* __restrict__ kern,
                                 const float* __restrict__ invnorm,
                                 float* __restrict__ y,
                                 float* __restrict__ th) {
    const int lane = threadIdx.x & 31;
    const int wave = threadIdx.x >> 5;
    const int row  = blockIdx.x * 8 + wave;
    const float* xr = X + (size_t)row * F_DIM;
    float s = 0.f;
    #pragma unroll
    for (int f = lane; f < F_DIM; f += 32) s += xr[f] * kern[f];
    #pragma unroll
    for (int o = 16; o > 0; o >>= 1) s += __shfl_xor(s, o, 32);
    if (lane == 0) {
        float yy = s * invnorm[0];
        y[row]  = yy;
        th[row] = tanhf(yy);
    }
}

// ---- kernel 2: rank each score; keep if rank < K_KEEP (stable tie-break) ---
__global__ void tkp_rank_kernel(const float* __restrict__ y,
                                int* __restrict__ keep) {
    __shared__ float sy[N_NODES];
    for (int j = threadIdx.x; j < N_NODES; j += blockDim.x) sy[j] = y[j];
    __syncthreads();
    const int i = blockIdx.x * blockDim.x + threadIdx.x;
    const float yi = sy[i];
    int cnt = 0;
    for (int j = 0; j < N_NODES; ++j) {
        float yj = sy[j];
        cnt += (yj > yi) || ((yj == yi) && (j < i));
    }
    keep[i] = (cnt < K_KEEP) ? 1 : 0;
}

// ---- kernel 3: ordered compaction -> sorted idx (single wave) --------------
__global__ void tkp_compact_kernel(const int* __restrict__ keep,
                                   int* __restrict__ idx) {
    const int lane = threadIdx.x;
    int total = 0;
    for (int base = 0; base < N_NODES; base += 32) {
        int f = keep[base + lane];
        unsigned long long m = __ballot(f != 0);
        int pos = total + __popcll(m & ((1ull << lane) - 1ull));
        if (f) idx[pos] = base + lane;
        total += __popcll(m);
    }
}

// ---- kernel 4: X_pooled[r,:] = X[idx[r],:] * tanh(y[idx[r]]) ---------------
__global__ void tkp_xpool_kernel(const float* __restrict__ X,
                                 const int* __restrict__ idx,
                                 const float* __restrict__ th,
                                 float* __restrict__ outX) {
    const int r = blockIdx.x;
    const int i = idx[r];
    const float t = th[i];
    outX[(size_t)r * F_DIM + threadIdx.x] = X[(size_t)i * F_DIM + threadIdx.x] * t;
}

// ---- kernel 5: S_pooled[r] = S[idx[r]] (emitted as float) ------------------
__global__ void tkp_spool_kernel(const int* __restrict__ S,
                                 const int* __restrict__ idx,
                                 float* __restrict__ outS) {
    const int r = blockIdx.x * blockDim.x + threadIdx.x;
    if (r < K_KEEP) outS[r] = (float)S[idx[r]];
}

// ---- kernel 6a: gathered transpose  BT[c,k] = A[k, idx[c]] -----------------
// 32x32 tiles via LDS (padded to kill bank conflicts); block (32,8).
__global__ void tkp_gatherT_kernel(const float* __restrict__ A,
                                   const int* __restrict__ idx,
                                   float* __restrict__ BT) {
    __shared__ float tile[32][33];
    const int k0 = blockIdx.x * 32;
    const int c0 = blockIdx.y * 32;
    const int tx = threadIdx.x;          // 0..31
    const int ty = threadIdx.y;          // 0..7
    const int gcol = idx[c0 + tx];       // gathered column for this lane
    #pragma unroll
    for (int r = 0; r < 4; ++r) {
        const int kk = ty + r * 8;
        tile[kk][tx] = A[(size_t)(k0 + kk) * N_NODES + gcol];
    }
    __syncthreads();
    #pragma unroll
    for (int r = 0; r < 4; ++r) {
        const int cc = ty + r * 8;
        BT[(size_t)(c0 + cc) * N_NODES + k0 + tx] = tile[tx][cc];
    }
}

// ---- kernel 6b: GEMM with pre-transposed B (all frag loads are b64) --------
__global__ __launch_bounds__(128, 1)
void tkp_gemm_pool_bt_kernel(const float* __restrict__ A,
                             const float* __restrict__ BT,
                             const int* __restrict__ idx,
                             float* __restrict__ outA) {
    const int lane   = threadIdx.x & 31;
    const int wave   = threadIdx.x >> 5;
    const int waveM  = wave >> 1;
    const int waveN  = wave & 1;
    const int rowBase = blockIdx.y * 128 + waveM * 64;
    const int colBase = blockIdx.x * 128 + waveN * 64;
    const int laneM   = lane & 15;   // M (A frag) / N (B frag)
    const int laneKhi = lane >> 4;   // high K pair selector

    const float* pa[4];
    const float* pb[4];
    #pragma unroll
    for (int m = 0; m < 4; ++m) {
        int r = idx[rowBase + m * 16 + laneM];
        pa[m] = A + (size_t)r * N_NODES + (size_t)(laneKhi * 2);
    }
    #pragma unroll
    for (int n = 0; n < 4; ++n) {
        pb[n] = BT + (size_t)(colBase + n * 16 + laneM) * N_NODES
                   + (size_t)(laneKhi * 2);
    }

    v8f acc[4][4];
    #pragma unroll
    for (int m = 0; m < 4; ++m)
        #pragma unroll
        for (int n = 0; n < 4; ++n)
            acc[m][n] = (v8f)(0.0f);

    for (int k = 0; k < N_NODES; k += 4) {
        v2f a[4], b[4];
        #pragma unroll
        for (int m = 0; m < 4; ++m) a[m] = *(const v2f*)(pa[m] + k);
        #pragma unroll
        for (int n = 0; n < 4; ++n) b[n] = *(const v2f*)(pb[n] + k);
        #pragma unroll
        for (int m = 0; m < 4; ++m)
            #pragma unroll
            for (int n = 0; n < 4; ++n)
                acc[m][n] = __builtin_amdgcn_wmma_f32_16x16x4_f32(
                    false, a[m], false, b[n], (short)0, acc[m][n], false, false);
    }

    #pragma unroll
    for (int m = 0; m < 4; ++m) {
        #pragma unroll
        for (int n = 0; n < 4; ++n) {
            const int col = colBase + n * 16 + laneM;
            #pragma unroll
            for (int v = 0; v < 8; ++v) {
                const int row = rowBase + m * 16 + v + 8 * laneKhi;
                outA[(size_t)row * K_KEEP + col] = acc[m][n][v];
            }
        }
    }
}

// ---- kernel 6c: fallback GEMM, direct (strided) B gather -------------------
__global__ __launch_bounds__(128, 1)
void tkp_gemm_pool_kernel(const float* __restrict__ A,
                          const int* __restrict__ idx,
                          float* __restrict__ outA) {
    const int lane   = threadIdx.x & 31;
    const int wave   = threadIdx.x >> 5;
    const int waveM  = wave >> 1;
    const int waveN  = wave & 1;
    const int rowBase = blockIdx.y * 128 + waveM * 64;
    const int colBase = blockIdx.x * 128 + waveN * 64;
    const int laneM   = lane & 15;
    const int laneKhi = lane >> 4;

    const float* pa[4];
    int colI[4];
    #pragma unroll
    for (int m = 0; m < 4; ++m) {
        int r = idx[rowBase + m * 16 + laneM];
        pa[m] = A + (size_t)r * N_NODES + (size_t)(laneKhi * 2);
    }
    #pragma unroll
    for (int n = 0; n < 4; ++n)
        colI[n] = idx[colBase + n * 16 + laneM];

    v8f acc[4][4];
    #pragma unroll
    for (int m = 0; m < 4; ++m)
        #pragma unroll
        for (int n = 0; n < 4; ++n)
            acc[m][n] = (v8f)(0.0f);

    for (int k = 0; k < N_NODES; k += 4) {
        v2f a[4], b[4];
        const int kk = k + laneKhi * 2;
        #pragma unroll
        for (int m = 0; m < 4; ++m)
            a[m] = *(const v2f*)(pa[m] + k);
        #pragma unroll
        for (int n = 0; n < 4; ++n) {
            b[n].x = A[(size_t)kk * N_NODES + colI[n]];
            b[n].y = A[(size_t)(kk + 1) * N_NODES + colI[n]];
        }
        #pragma unroll
        for (int m = 0; m < 4; ++m)
            #pragma unroll
            for (int n = 0; n < 4; ++n)
                acc[m][n] = __builtin_amdgcn_wmma_f32_16x16x4_f32(
                    false, a[m], false, b[n], (short)0, acc[m][n], false, false);
    }

    #pragma unroll
    for (int m = 0; m < 4; ++m) {
        #pragma unroll
        for (int n = 0; n < 4; ++n) {
            const int col = colBase + n * 16 + laneM;
            #pragma unroll
            for (int v = 0; v < 8; ++v) {
                const int row = rowBase + m * 16 + v + 8 * laneKhi;
                outA[(size_t)row * K_KEEP + col] = acc[m][n][v];
            }
        }
    }
}

// ---------------------------------------------------------------------------
extern "C" void kernel_launch(void* const* d_in, const int* in_sizes, int n_in,
                              void* d_out, int out_size, void* d_ws, size_t ws_size,
                              hipStream_t stream) {
    const float* X    = (const float*)d_in[0];   // (4096, 256)
    const float* A    = (const float*)d_in[1];   // (4096, 4096)
    const int*   S    = (const int*)d_in[2];     // (4096,)
    const float* kern = (const float*)d_in[3];   // (256, 1)

    // workspace carve-out (small scratch first, then optional 32 MB BT)
    float* invnorm = (float*)d_ws;                       // 1 float (+pad)
    float* y       = invnorm + 16;                       // 4096
    float* th      = y + N_NODES;                        // 4096
    int*   keep    = (int*)(th + N_NODES);               // 4096
    int*   idx     = keep + N_NODES;                     // 2048
    // align BT to 256B past the small scratch
    size_t small_bytes = ((size_t)(16 + 2 * N_NODES) * 4 +
                          (size_t)(N_NODES + K_KEEP) * 4 + 255) & ~(size_t)255;
    float* BT = (float*)((char*)d_ws + small_bytes);     // (2048, 4096) = 32 MB
    size_t bt_bytes = (size_t)K_KEEP * N_NODES * sizeof(float);
    const bool use_bt = (ws_size >= small_bytes + bt_bytes);

    // outputs, concatenated flat in return order
    float* outX = (float*)d_out;                         // 2048*256
    float* outA = outX + (size_t)K_KEEP * F_DIM;         // 2048*2048
    float* outS = outA + (size_t)K_KEEP * K_KEEP;        // 2048

    tkp_norm_kernel<<<1, 256, 0, stream>>>(kern, invnorm);
    tkp_score_kernel<<<N_NODES / 8, 256, 0, stream>>>(X, kern, invnorm, y, th);
    tkp_rank_kernel<<<N_NODES / 256, 256, 0, stream>>>(y, keep);
    tkp_compact_kernel<<<1, 32, 0, stream>>>(keep, idx);
    tkp_xpool_kernel<<<K_KEEP, F_DIM, 0, stream>>>(X, idx, th, outX);
    tkp_spool_kernel<<<K_KEEP / 256, 256, 0, stream>>>(S, idx, outS);

    dim3 ggrid(K_KEEP / 128, K_KEEP / 128);  // 16 x 16 blocks of 128 threads
    if (use_bt) {
        dim3 tgrid(N_NODES / 32, K_KEEP / 32);
        dim3 tblk(32, 8);
        tkp_gatherT_kernel<<<tgrid, tblk, 0, stream>>>(A, idx, BT);
        tkp_gemm_pool_bt_kernel<<<ggrid, 128, 0, stream>>>(A, BT, idx, outA);
    } else {
        tkp_gemm_pool_kernel<<<ggrid, 128, 0, stream>>>(A, idx, outA);
    }
}